// FastGTConv_19413252178073
// MI455X (gfx1250) — compile-verified
//
#include <hip/hip_runtime.h>
#include <hip/hip_bf16.h>

// ---------------- Problem constants (from reference) ----------------
constexpr int KT = 5;                 // K_TYPES
constexpr int EP = 2000000;           // E_PER
constexpr int KE = KT * EP;           // 10,000,000 total edges
constexpr int CO = 4;                 // C_OUT
// N_NODES = 50,000 < 2^16  ==>  (row<<16)|col fits in 32 bits.

// ---------------- Radix sort config (32-bit keys, 4 x 8-bit passes) -----
constexpr int RS_BLOCK = 256;
constexpr int RS_ITEMS = 8;
constexpr int RS_TILE  = RS_BLOCK * RS_ITEMS;               // 2048
constexpr int RS_NBLK  = (KE + RS_TILE - 1) / RS_TILE;      // 4883
constexpr int HIST_N   = 256 * RS_NBLK;                     // 1,250,048

// ---------------- Hierarchical scan config ----------------
constexpr int SC_BLOCK = 256;
constexpr int SC_ITEMS = 16;
constexpr int SC_TILE  = SC_BLOCK * SC_ITEMS;               // 4096
constexpr int SC_NB_HIST = (HIST_N + SC_TILE - 1) / SC_TILE; // 306

// ---------------- Segment (dedup) config ----------------
constexpr int SG_BLOCK = 256;
constexpr int SG_ITEMS = 16;
constexpr int SG_TILE  = SG_BLOCK * SG_ITEMS;               // 4096
constexpr int SG_NBLK  = (KE + SG_TILE - 1) / SG_TILE;      // 2442

typedef unsigned u32x4 __attribute__((ext_vector_type(4)));
typedef unsigned u32x8 __attribute__((ext_vector_type(8)));

// ---------------- CDNA5 async DMA: global -> LDS (ASYNCcnt-tracked) -----
__device__ __forceinline__ void async_copy_u32(unsigned* lds_dst, const unsigned* gsrc) {
  // Low 32 bits of the generic (flat) pointer are the LDS byte offset.
  unsigned lds_off = (unsigned)(unsigned long long)lds_dst;
  unsigned long long ga = (unsigned long long)gsrc;
  asm volatile("global_load_async_to_lds_b32 %0, %1, off"
               :: "v"(lds_off), "v"(ga) : "memory");
}
__device__ __forceinline__ void wait_asynccnt0() {
  asm volatile("s_wait_asynccnt 0x0" ::: "memory");
}

// ---------------- CDNA5 Tensor Data Mover: one-descriptor tile DMA ------
// Loads a contiguous RS_TILE x u32 (8KB) block from global into LDS with a
// single TENSOR_LOAD_TO_LDS. D# built per CDNA5 ISA:
//   group0: count=1 | lds_addr | global_addr[56:0] | type=2
//   group1: data_size=2 (4B), tensor_dim0=tile_dim0=2048,
//           tensor_dim1=tile_dim1=1, dim0_stride=2048, no pad/multicast.
// Issue from ONE wave per block (TDM issues per wave, EXEC ignored),
// wait TENSORcnt on that wave, then __syncthreads() publishes the tile.
__device__ __forceinline__ void tdm_load_tile_2048(unsigned* lds_dst, const unsigned* gsrc) {
  unsigned lds_off = (unsigned)(unsigned long long)lds_dst;
  unsigned long long ga = (unsigned long long)gsrc;
  u32x4 g0;
  g0[0] = 1u;                                             // count=1 (valid), user D#
  g0[1] = lds_off;                                        // lds_addr (bytes)
  g0[2] = (unsigned)ga;                                   // global_addr[31:0]
  g0[3] = (unsigned)((ga >> 32) & 0x01FFFFFFull)          // global_addr[56:32]
          | (2u << 30);                                   // type=2 ("image")
  u32x8 g1;
  g1[0] = (2u << 16);                                     // wg_mask=0, data_size=2 (4B)
  g1[1] = (unsigned)(RS_TILE & 0xFFFF) << 16;             // tensor_dim0[15:0]
  g1[2] = (unsigned)(RS_TILE >> 16) | (1u << 16);         // tensor_dim0[31:16] | tensor_dim1=1
  g1[3] = ((unsigned)RS_TILE) << 16;                      // tile_dim0=2048
  g1[4] = 1u;                                             // tile_dim1=1, tile_dim2=0
  g1[5] = (unsigned)RS_TILE;                              // tensor_dim0_stride[31:0]
  g1[6] = 0u;                                             // stride0[47:32], stride1[15:0]
  g1[7] = 0u;                                             // stride1[47:16]
  asm volatile("tensor_load_to_lds %0, %1"
               :: "s"(g0), "s"(g1) : "memory");
}
__device__ __forceinline__ void wait_tensorcnt0() {
#if __has_builtin(__builtin_amdgcn_s_wait_tensorcnt)
  __builtin_amdgcn_s_wait_tensorcnt(0);
#else
  asm volatile("s_wait_tensorcnt 0x0" ::: "memory");
#endif
}

// ---------------- Kernels ----------------

// Zero the out_index + out_vals regions (6*KE dwords).
__global__ void zero_u32_kernel(unsigned* __restrict__ p, long long n) {
  long long i = (long long)blockIdx.x * blockDim.x + threadIdx.x;
  long long stride = (long long)gridDim.x * blockDim.x;
  for (; i < n; i += stride) p[i] = 0u;
}

// filt = softmax(weight, axis=1); write to workspace and to d_out tail.
__global__ void softmax_filt_kernel(const float* __restrict__ w,
                                    float* __restrict__ filt_ws,
                                    float* __restrict__ filt_out) {
  int c = threadIdx.x;
  if (c < CO) {
    float m = w[c * KT];
    for (int k = 1; k < KT; ++k) m = fmaxf(m, w[c * KT + k]);
    float e[KT]; float s = 0.f;
    for (int k = 0; k < KT; ++k) { e[k] = __expf(w[c * KT + k] - m); s += e[k]; }
    float inv = 1.f / s;
    for (int k = 0; k < KT; ++k) {
      float v = e[k] * inv;
      filt_ws[c * KT + k]  = v;
      filt_out[c * KT + k] = v;
    }
  }
}

// Build keys (row<<16)|col + payload, fused with the pass-0 digit histogram
// (saves one full 40MB read pass over the key array).
__global__ void build_keys_hist_kernel(const int* __restrict__ eidx,
                                       unsigned* __restrict__ keys,
                                       unsigned* __restrict__ pay,
                                       unsigned* __restrict__ hist) {
  __shared__ unsigned bins[256];
  int t = threadIdx.x, b = blockIdx.x;
  long long base = (long long)b * RS_TILE;
  bins[t] = 0u;
  __syncthreads();
#pragma unroll
  for (int i = 0; i < RS_ITEMS; ++i) {
    long long p = base + i * RS_BLOCK + t;
    if (p < KE) {
      int k = (int)(p / EP);
      int e = (int)(p - (long long)k * EP);
      const int* rbase = eidx + (long long)(k * 2 + 0) * EP;
      const int* cbase = eidx + (long long)(k * 2 + 1) * EP;
      __builtin_prefetch(rbase + e + RS_TILE, 0, 0);   // gfx1250 global_prefetch_b8
      unsigned r = (unsigned)rbase[e];
      unsigned c = (unsigned)cbase[e];
      unsigned key = (r << 16) | (c & 0xFFFFu);
      keys[p] = key;
      pay[p]  = (unsigned)p;
      atomicAdd(&bins[key & 0xFFu], 1u);
    }
  }
  __syncthreads();
  hist[t * RS_NBLK + b] = bins[t];   // digit-major for direct global scan
}

// Per-block 256-bin digit histogram (passes 1..3).
// Full tiles: one TDM descriptor DMA. Tail tile: per-lane async loads.
__global__ void radix_hist_kernel(const unsigned* __restrict__ keys,
                                  unsigned* __restrict__ hist, int shift) {
  __shared__ unsigned lkeys[RS_TILE];
  __shared__ unsigned bins[256];
  int t = threadIdx.x, b = blockIdx.x;
  long long base = (long long)b * RS_TILE;
  bins[t] = 0u;
  if (base + RS_TILE <= KE) {
    if (t < 32) {                       // wave 0 issues a single tile DMA
      tdm_load_tile_2048(lkeys, &keys[base]);
      wait_tensorcnt0();
    }
  } else {
#pragma unroll
    for (int i = 0; i < RS_ITEMS; ++i) {
      long long p = base + i * RS_BLOCK + t;
      if (p < KE) async_copy_u32(&lkeys[i * RS_BLOCK + t], &keys[p]);
    }
    wait_asynccnt0();
  }
  __syncthreads();
#pragma unroll
  for (int i = 0; i < RS_ITEMS; ++i) {
    long long p = base + i * RS_BLOCK + t;
    if (p < KE) {
      unsigned d = (lkeys[i * RS_BLOCK + t] >> shift) & 0xFFu;
      atomicAdd(&bins[d], 1u);
    }
  }
  __syncthreads();
  hist[t * RS_NBLK + b] = bins[t];
}

// Hierarchical scan, level 1: per-tile exclusive scan; tile totals -> aux.
__global__ void scan_local_kernel(unsigned* __restrict__ data, int n,
                                  unsigned* __restrict__ aux) {
  __shared__ unsigned tmp[SC_BLOCK];
  int t = threadIdx.x;
  int tb = blockIdx.x * SC_TILE + t * SC_ITEMS;
  unsigned vals[SC_ITEMS];
  unsigned s = 0u;
#pragma unroll
  for (int i = 0; i < SC_ITEMS; ++i) {
    int p = tb + i;
    vals[i] = (p < n) ? data[p] : 0u;
    s += vals[i];
  }
  tmp[t] = s; __syncthreads();
  for (int off = 1; off < SC_BLOCK; off <<= 1) {
    unsigned x = (t >= off) ? tmp[t - off] : 0u;
    __syncthreads();
    tmp[t] += x;
    __syncthreads();
  }
  unsigned run = tmp[t] - s;   // exclusive prefix within tile
#pragma unroll
  for (int i = 0; i < SC_ITEMS; ++i) {
    int p = tb + i;
    if (p < n) data[p] = run;
    run += vals[i];
  }
  if (t == SC_BLOCK - 1) aux[blockIdx.x] = tmp[SC_BLOCK - 1];
}

// Single-block looping exclusive scan (aux arrays / block sums).
__global__ void scan_single_kernel(unsigned* __restrict__ data, int n) {
  __shared__ unsigned tmp[1024];
  int t = threadIdx.x;
  unsigned carry = 0u;
  for (int base = 0; base < n; base += 1024) {
    int i = base + t;
    unsigned v = (i < n) ? data[i] : 0u;
    tmp[t] = v; __syncthreads();
    for (int off = 1; off < 1024; off <<= 1) {
      unsigned x = (t >= off) ? tmp[t - off] : 0u;
      __syncthreads();
      tmp[t] += x;
      __syncthreads();
    }
    unsigned incl  = tmp[t];
    unsigned total = tmp[1023];
    if (i < n) data[i] = incl - v + carry;
    carry += total;              // identical in all threads
    __syncthreads();
  }
}

// Hierarchical scan, level 3: add tile offsets back.
__global__ void scan_add_kernel(unsigned* __restrict__ data, int n,
                                const unsigned* __restrict__ aux) {
  int t = threadIdx.x, b = blockIdx.x;
  unsigned add = aux[b];
#pragma unroll
  for (int i = 0; i < SC_ITEMS; ++i) {
    int p = b * SC_TILE + i * SC_BLOCK + t;
    if (p < n) data[p] += add;
  }
}

// Stable scatter pass of LSD radix sort.
// Full tiles staged via two TDM descriptor DMAs (keys + payload); tail via
// per-lane async loads. wave32 ballot match-any ranking:
//   rank = popcount(same-digit peers below me) + earlier-wave digit counts.
__global__ void radix_scatter_kernel(const unsigned* __restrict__ keys_in,
                                     const unsigned* __restrict__ pay_in,
                                     unsigned* __restrict__ keys_out,
                                     unsigned* __restrict__ pay_out,
                                     const unsigned* __restrict__ hist, int shift) {
  __shared__ unsigned lkeys[RS_TILE];
  __shared__ unsigned lpay[RS_TILE];
  __shared__ unsigned gofs[256];
  __shared__ unsigned before[256];
  __shared__ unsigned wcnt[8 * 256];          // [wave][digit] per-round counts
  int t = threadIdx.x, b = blockIdx.x;
  int lane = t & 31, wid = t >> 5;
  long long base = (long long)b * RS_TILE;
  gofs[t]   = hist[t * RS_NBLK + b];
  before[t] = 0u;

  if (base + RS_TILE <= KE) {
    if (t < 32) {                       // wave 0: two descriptor DMAs, in order
      tdm_load_tile_2048(lkeys, &keys_in[base]);
      tdm_load_tile_2048(lpay,  &pay_in[base]);
      wait_tensorcnt0();
    }
  } else {
#pragma unroll
    for (int i = 0; i < RS_ITEMS; ++i) {
      long long p = base + i * RS_BLOCK + t;
      if (p < KE) {
        async_copy_u32(&lkeys[i * RS_BLOCK + t], &keys_in[p]);
        async_copy_u32(&lpay[i * RS_BLOCK + t],  &pay_in[p]);
      }
    }
    wait_asynccnt0();
  }
  __syncthreads();

  long long remll = KE - base;
  int rem = (remll < (long long)RS_TILE) ? (int)remll : RS_TILE;
  unsigned lanelt = (lane == 0) ? 0u : (0xFFFFFFFFu >> (32 - lane));

  for (int i = 0; i < RS_ITEMS; ++i) {
#pragma unroll
    for (int w2 = 0; w2 < 8; ++w2) wcnt[w2 * 256 + t] = 0u;   // zero counters
    __syncthreads();

    int li = i * RS_BLOCK + t;
    bool act = li < rem;
    unsigned key = 0u, pay = 0u, d = 0u;
    if (act) { key = lkeys[li]; pay = lpay[li]; d = (key >> shift) & 0xFFu; }

    // wave32 match-any on the 8-bit digit
    unsigned active = __builtin_amdgcn_ballot_w32(act);
    unsigned peers  = active;
#pragma unroll
    for (int bit = 0; bit < 8; ++bit) {
      unsigned bb = __builtin_amdgcn_ballot_w32(act && ((d >> bit) & 1u));
      peers &= ((d >> bit) & 1u) ? bb : ~bb;
    }
    unsigned rank = 0u;
    if (act) {
      unsigned below = peers & lanelt;
      rank = (unsigned)__builtin_popcount(below);
      if (below == 0u)                             // lowest same-digit lane leads
        wcnt[wid * 256 + d] = (unsigned)__builtin_popcount(peers);
    }
    __syncthreads();

    unsigned tot = 0u;                             // this round's total for digit t
#pragma unroll
    for (int w2 = 0; w2 < 8; ++w2) tot += wcnt[w2 * 256 + t];
    if (act) {
      unsigned cross = 0u;                         // earlier waves, same digit
      for (int w2 = 0; w2 < wid; ++w2) cross += wcnt[w2 * 256 + d];
      unsigned pos = gofs[d] + before[d] + cross + rank;
      keys_out[pos] = key;
      pay_out[pos]  = pay;
    }
    __syncthreads();
    before[t] += tot;
    __syncthreads();
  }
}

__device__ __forceinline__ bool seg_flag(const unsigned* keys, long long p) {
  return (p == 0) || (keys[p] != keys[p - 1]);
}

// Count "new segment" flags per tile.
__global__ void seg_block_counts_kernel(const unsigned* __restrict__ keys,
                                        unsigned* __restrict__ bs) {
  __shared__ unsigned red[SG_BLOCK];
  int t = threadIdx.x;
  long long base = (long long)blockIdx.x * SG_TILE;
  unsigned cnt = 0u;
#pragma unroll
  for (int i = 0; i < SG_ITEMS; ++i) {
    long long p = base + i * SG_BLOCK + t;
    if (p < KE) cnt += seg_flag(keys, p) ? 1u : 0u;
  }
  red[t] = cnt; __syncthreads();
  for (int s = SG_BLOCK / 2; s > 0; s >>= 1) {
    if (t < s) red[t] += red[t + s];
    __syncthreads();
  }
  if (t == 0) bs[blockIdx.x] = red[0];
}

// Write segment id per sorted edge + coalesced (row,col) at segment heads.
__global__ void seg_write_kernel(const unsigned* __restrict__ keys,
                                 const unsigned* __restrict__ bofs,
                                 unsigned* __restrict__ seg,
                                 int* __restrict__ out_index) {
  __shared__ unsigned tmp[SG_BLOCK];
  int t = threadIdx.x;
  long long tb = (long long)blockIdx.x * SG_TILE + (long long)t * SG_ITEMS;
  unsigned cnt = 0u;
#pragma unroll
  for (int i = 0; i < SG_ITEMS; ++i) {
    long long p = tb + i;
    if (p < KE) cnt += seg_flag(keys, p) ? 1u : 0u;
  }
  tmp[t] = cnt; __syncthreads();
  for (int off = 1; off < SG_BLOCK; off <<= 1) {
    unsigned x = (t >= off) ? tmp[t - off] : 0u;
    __syncthreads();
    tmp[t] += x;
    __syncthreads();
  }
  unsigned run = bofs[blockIdx.x] + (tmp[t] - cnt);  // flags strictly before tb
  for (int i = 0; i < SG_ITEMS; ++i) {
    long long p = tb + i;
    if (p >= KE) break;
    unsigned key = keys[p];
    bool f = (p == 0) || (keys[p - 1] != key);
    if (f) {
      seg[p] = run;
      out_index[run]                 = (int)(key >> 16);       // out_rows
      out_index[(long long)KE + run] = (int)(key & 0xFFFFu);   // out_cols
      ++run;
    } else {
      seg[p] = run - 1u;
    }
  }
}

// Per-segment weighted sums for all 4 channels (dup rate ~2e-3, segments tiny).
__global__ void segsum_kernel(const unsigned* __restrict__ keys,
                              const unsigned* __restrict__ pay,
                              const unsigned* __restrict__ seg,
                              const float* __restrict__ evals,
                              const float* __restrict__ filt,
                              float* __restrict__ out_vals) {
  __shared__ float sf[CO * KT];
  int t = threadIdx.x;
  if (t < CO * KT) sf[t] = filt[t];
  __syncthreads();
  long long p = (long long)blockIdx.x * blockDim.x + t;
  if (p >= KE) return;
  unsigned k0 = keys[p];
  if (p != 0 && keys[p - 1] == k0) return;     // not a segment head
  float a0 = 0.f, a1 = 0.f, a2 = 0.f, a3 = 0.f;
  long long q = p;
  do {
    unsigned id = pay[q];
    unsigned kt = id / (unsigned)EP;           // edge type 0..4
    float v = evals[id];
    a0 += sf[0 * KT + kt] * v;
    a1 += sf[1 * KT + kt] * v;
    a2 += sf[2 * KT + kt] * v;
    a3 += sf[3 * KT + kt] * v;
    ++q;
  } while (q < KE && keys[q] == k0);
  unsigned s = seg[p];
  out_vals[0LL * KE + s] = a0;
  out_vals[1LL * KE + s] = a1;
  out_vals[2LL * KE + s] = a2;
  out_vals[3LL * KE + s] = a3;
}

// ---------------- Host driver ----------------
extern "C" void kernel_launch(void* const* d_in, const int* in_sizes, int n_in,
                              void* d_out, int out_size, void* d_ws, size_t ws_size,
                              hipStream_t stream) {
  (void)in_sizes; (void)n_in; (void)out_size; (void)ws_size;
  const int*   eidx  = (const int*)d_in[0];     // [5,2,2M] int32
  const float* evals = (const float*)d_in[1];   // [5,2M] f32
  const float* w     = (const float*)d_in[2];   // [4,5]  f32
  // d_in[3] = num_nodes (unused; N < 2^16 is baked into the key packing)

  // Output layout: out_index [2*KE] int32, out_vals [4*KE] f32, filt [20] f32.
  float* outF      = (float*)d_out;
  int*   out_index = (int*)d_out;
  float* out_vals  = outF + 2LL * KE;
  float* out_filt  = outF + 6LL * KE;

  // Workspace layout (~165 MB).
  unsigned* keysA = (unsigned*)d_ws;
  unsigned* idxA  = keysA + KE;
  unsigned* keysB = idxA + KE;
  unsigned* idxB  = keysB + KE;
  unsigned* hist  = idxB + KE;            // HIST_N
  unsigned* aux   = hist + HIST_N;        // >= SC_NB_HIST (pad 512)
  unsigned* bsum  = aux + 512;            // >= SG_NBLK (pad 2560)
  float* filt_ws  = (float*)(bsum + 2560);
  unsigned* segA  = keysB;                // reuse after sort finishes in A

  zero_u32_kernel<<<4096, 256, 0, stream>>>((unsigned*)d_out, 6LL * KE);
  softmax_filt_kernel<<<1, 32, 0, stream>>>(w, filt_ws, out_filt);
  // Fused: build keys/payload + pass-0 histogram.
  build_keys_hist_kernel<<<RS_NBLK, 256, 0, stream>>>(eidx, keysA, idxA, hist);

  for (int pass = 0; pass < 4; ++pass) {
    const unsigned* sk = (pass & 1) ? keysB : keysA;
    const unsigned* sp = (pass & 1) ? idxB  : idxA;
    unsigned* dk = (pass & 1) ? keysA : keysB;
    unsigned* dp = (pass & 1) ? idxA  : idxB;
    int shift = pass * 8;
    if (pass > 0)
      radix_hist_kernel<<<RS_NBLK, 256, 0, stream>>>(sk, hist, shift);
    scan_local_kernel<<<SC_NB_HIST, SC_BLOCK, 0, stream>>>(hist, HIST_N, aux);
    scan_single_kernel<<<1, 1024, 0, stream>>>(aux, SC_NB_HIST);
    scan_add_kernel<<<SC_NB_HIST, SC_BLOCK, 0, stream>>>(hist, HIST_N, aux);
    radix_scatter_kernel<<<RS_NBLK, 256, 0, stream>>>(sk, sp, dk, dp, hist, shift);
  }
  // After 4 passes sorted data is in keysA/idxA; keysB/idxB are free.

  seg_block_counts_kernel<<<SG_NBLK, SG_BLOCK, 0, stream>>>(keysA, bsum);
  scan_single_kernel<<<1, 1024, 0, stream>>>(bsum, SG_NBLK);
  seg_write_kernel<<<SG_NBLK, SG_BLOCK, 0, stream>>>(keysA, bsum, segA, out_index);
  segsum_kernel<<<(KE + 255) / 256, 256, 0, stream>>>(keysA, idxA, segA,
                                                      evals, filt_ws, out_vals);
}